// Base2LayerGNN_32547262169571
// MI455X (gfx1250) — compile-verified
//
#include <hip/hip_runtime.h>

typedef __attribute__((ext_vector_type(2))) float v2f;
typedef __attribute__((ext_vector_type(8))) float v8f;

#define FEAT 64

// ---------------- zero fill ----------------
__global__ void zero_f32(float* __restrict__ p, long n) {
    long i = (long)blockIdx.x * blockDim.x + threadIdx.x;
    long stride = (long)gridDim.x * blockDim.x;
    for (; i < n; i += stride) p[i] = 0.0f;
}

// ---------------- degree (float so the divide path matches reference) ----------------
__global__ void degree_kernel(const int* __restrict__ dst, float* __restrict__ deg, int E) {
    int e = blockIdx.x * blockDim.x + threadIdx.x;
    if (e < E) atomicAdd(&deg[dst[e]], 1.0f);
}

// ---------------- edge scatter: aggr[dst] += x[src]  (16 threads / edge, float4 each) ----------------
__global__ void scatter_kernel(const float* __restrict__ x,
                               const int* __restrict__ src,
                               const int* __restrict__ dst,
                               float* __restrict__ aggr, long E) {
    long t = (long)blockIdx.x * blockDim.x + threadIdx.x;
    if (t >= E * 16) return;
    long e = t >> 4;
    int  f = (int)(t & 15) << 2;           // feature offset, multiple of 4 -> 16B aligned
    long s = src[e];
    long d = dst[e];
    const float4 v = *(const float4*)(x + s * FEAT + f);
    float* a = aggr + d * FEAT + f;
    atomicAdd(a + 0, v.x);
    atomicAdd(a + 1, v.y);
    atomicAdd(a + 2, v.z);
    atomicAdd(a + 3, v.w);
}

// ---------------- fused SAGE update:
//   out[16 nodes x 64] = (aggr/deg) @ Wl^T + x @ Wr^T + bl
// one wave per 16-node tile, V_WMMA_F32_16X16X4_F32, both GEMMs share one accumulator.
__global__ void __launch_bounds__(256)
sage_update_wmma(const float* __restrict__ x, const float* __restrict__ aggr,
                 const float* __restrict__ deg,
                 const float* __restrict__ Wl, const float* __restrict__ bl,
                 const float* __restrict__ Wr,
                 float* __restrict__ out, int N) {
    __shared__ float Wl_s[FEAT * FEAT];
    __shared__ float Wr_s[FEAT * FEAT];
    __shared__ float b_s[FEAT];

    // cooperative stage of weights + bias into LDS (float4 granularity)
    for (int i = threadIdx.x; i < FEAT * FEAT / 4; i += 256) {
        ((float4*)Wl_s)[i] = ((const float4*)Wl)[i];
        ((float4*)Wr_s)[i] = ((const float4*)Wr)[i];
    }
    if (threadIdx.x < FEAT) b_s[threadIdx.x] = bl[threadIdx.x];
    __syncthreads();

    const int lane = threadIdx.x & 31;
    const int wave = threadIdx.x >> 5;
    const int tile = blockIdx.x * 8 + wave;     // 16-node tile id (wave-uniform)
    const int base = tile * 16;
    if (base >= N) return;                      // uniform per wave: EXEC stays all-ones

    const int row  = lane & 15;                 // M index (A/C) and N index (B/C)
    const int half = lane >> 4;                 // K-pair selector per ISA 16x4 layout
    const int node = base + row;
    const float dinv = 1.0f / fmaxf(deg[node], 1.0f);

    // Preload A fragments for both GEMMs: lane holds A[row][k0], A[row][k0+1]
    v2f a_ag[16], a_x[16];
    const float* agp = aggr + (long)node * FEAT;
    const float* xp  = x    + (long)node * FEAT;
#pragma unroll
    for (int ks = 0; ks < 16; ++ks) {
        const int k0 = ks * 4 + half * 2;
        v2f t0; t0.x = agp[k0] * dinv; t0.y = agp[k0 + 1] * dinv;
        a_ag[ks] = t0;
        v2f t1; t1.x = xp[k0];         t1.y = xp[k0 + 1];
        a_x[ks] = t1;
    }

#pragma unroll
    for (int nb = 0; nb < 4; ++nb) {
        const int ncol = nb * 16 + row;         // output column this lane supplies B for
        v8f c = {};
#pragma unroll
        for (int ks = 0; ks < 16; ++ks) {
            const int k0 = ks * 4 + half * 2;
            // B[k][n] = W[n][k]  (out = A @ W^T)
            v2f bfl; bfl.x = Wl_s[ncol * FEAT + k0]; bfl.y = Wl_s[ncol * FEAT + k0 + 1];
            c = __builtin_amdgcn_wmma_f32_16x16x4_f32(false, a_ag[ks], false, bfl,
                                                      (short)0, c, false, false);
            v2f bfr; bfr.x = Wr_s[ncol * FEAT + k0]; bfr.y = Wr_s[ncol * FEAT + k0 + 1];
            c = __builtin_amdgcn_wmma_f32_16x16x4_f32(false, a_x[ks], false, bfr,
                                                      (short)0, c, false, false);
        }
        const float bias = b_s[nb * 16 + row];
        // C layout: VGPR r -> M = r + half*8, N = lane&15
#pragma unroll
        for (int r = 0; r < 8; ++r) {
            const int M = r + half * 8;
            out[(long)(base + M) * FEAT + nb * 16 + row] = c[r] + bias;
        }
    }
}

extern "C" void kernel_launch(void* const* d_in, const int* in_sizes, int n_in,
                              void* d_out, int out_size, void* d_ws, size_t ws_size,
                              hipStream_t stream) {
    const float* x   = (const float*)d_in[0];
    const int*   eix = (const int*)d_in[1];     // edge_index, [2, E] flat (per-harness int32)
    const float* W1l = (const float*)d_in[2];
    const float* b1l = (const float*)d_in[3];
    const float* W1r = (const float*)d_in[4];
    const float* W2l = (const float*)d_in[5];
    const float* b2l = (const float*)d_in[6];
    const float* W2r = (const float*)d_in[7];

    const int  N = in_sizes[0] / FEAT;
    const long E = in_sizes[1] / 2;
    const int* src = eix;
    const int* dst = eix + E;

    float* deg  = (float*)d_ws;                 // N floats
    float* aggr = deg + N;                      // N*FEAT floats
    float* x1   = aggr + (size_t)N * FEAT;      // N*FEAT floats
    float* outp = (float*)d_out;

    const int deg_blocks = (int)((E + 255) / 256);
    const int sc_blocks  = (int)((E * 16 + 255) / 256);
    const int up_blocks  = (N + 127) / 128;     // 8 waves x 16 nodes per block

    // degrees are identical for both layers: compute once
    zero_f32<<<256, 256, 0, stream>>>(deg, (long)N);
    degree_kernel<<<deg_blocks, 256, 0, stream>>>(dst, deg, (int)E);

    // ---- layer 1 ----
    zero_f32<<<2048, 256, 0, stream>>>(aggr, (long)N * FEAT);
    scatter_kernel<<<sc_blocks, 256, 0, stream>>>(x, src, dst, aggr, E);
    sage_update_wmma<<<up_blocks, 256, 0, stream>>>(x, aggr, deg, W1l, b1l, W1r, x1, N);

    // ---- layer 2 ----
    zero_f32<<<2048, 256, 0, stream>>>(aggr, (long)N * FEAT);
    scatter_kernel<<<sc_blocks, 256, 0, stream>>>(x1, src, dst, aggr, E);
    sage_update_wmma<<<up_blocks, 256, 0, stream>>>(x1, aggr, deg, W2l, b2l, W2r, outp, N);
}